// SpatialAttention_wo_skip_25692494365577
// MI455X (gfx1250) — compile-verified
//
#include <hip/hip_runtime.h>

typedef __attribute__((ext_vector_type(2))) float v2f;
typedef __attribute__((ext_vector_type(4))) float v4f;
typedef __attribute__((ext_vector_type(8))) float v8f;

#define NPIX 4096
#define NBLK 1024   // NPIX/4
#define LOG2E 1.4426950408889634f

// D = A(16x4) * B(4x16) + C(16x16), all f32. Wave32, EXEC all-ones.
__device__ __forceinline__ v8f wmma4(v2f a, v2f b, v8f c) {
  return __builtin_amdgcn_wmma_f32_16x16x4_f32(
      /*neg_a=*/false, a, /*neg_b=*/false, b,
      /*c_mod=*/(short)0, c, /*reuse_a=*/false, /*reuse_b=*/false);
}

// ---------------------------------------------------------------------------
// Kernel 1: 1x1-conv projections.
//   v = W1*x + b1  -> vA packed as [b][n/4][c][n%4]   (WMMA-A operand layout)
//   q = W2*x + b2  -> qT [b][n][8]
//   k = W3*x + b3  -> kT [b][n][8]
// grid = B*16 blocks of 256 threads; each thread owns one pixel n.
// ---------------------------------------------------------------------------
__global__ __launch_bounds__(256) void proj_kernel(
    const float* __restrict__ x,
    const float* __restrict__ W1, const float* __restrict__ b1,
    const float* __restrict__ W2, const float* __restrict__ b2,
    const float* __restrict__ W3, const float* __restrict__ b3,
    float* __restrict__ vA, float* __restrict__ qT, float* __restrict__ kT)
{
  __shared__ float Wt1[64 * 64];      // [c][o]  (transposed for contiguous o)
  __shared__ float Wt23[64 * 16];     // [c][0:8]=W2 row, [c][8:16]=W3 row
  __shared__ float bias[80];          // b1[64], b2[8], b3[8]
  const int tid = threadIdx.x;

  for (int i = tid; i < 64 * 64; i += 256) {
    int o = i >> 6, c = i & 63;
    Wt1[c * 64 + o] = W1[i];
  }
  for (int i = tid; i < 8 * 64; i += 256) {
    int o = i >> 6, c = i & 63;
    Wt23[c * 16 + o]     = W2[i];
    Wt23[c * 16 + 8 + o] = W3[i];
  }
  if (tid < 64)      bias[tid] = b1[tid];
  else if (tid < 72) bias[tid] = b2[tid - 64];
  else if (tid < 80) bias[tid] = b3[tid - 72];
  __syncthreads();

  const int b = blockIdx.x >> 4;
  const int n = ((blockIdx.x & 15) << 8) + tid;

  float vacc[64], qacc[8], kacc[8];
#pragma unroll
  for (int o = 0; o < 64; ++o) vacc[o] = bias[o];
#pragma unroll
  for (int o = 0; o < 8; ++o) { qacc[o] = bias[64 + o]; kacc[o] = bias[72 + o]; }

  const float* xb = x + (size_t)b * 64 * NPIX + n;
  for (int c = 0; c < 64; ++c) {
    float xc = xb[(size_t)c * NPIX];
    const float* w1r = &Wt1[c * 64];
#pragma unroll
    for (int o = 0; o < 64; ++o) vacc[o] += w1r[o] * xc;
    const float* w2r = &Wt23[c * 16];
#pragma unroll
    for (int o = 0; o < 8; ++o) qacc[o] += w2r[o] * xc;
#pragma unroll
    for (int o = 0; o < 8; ++o) kacc[o] += w2r[8 + o] * xc;
  }

  // vA[((b*1024 + n/4)*64 + c)*4 + (n%4)]
  float* vout = vA + ((size_t)b * NBLK + (n >> 2)) * 256 + (n & 3);
#pragma unroll
  for (int o = 0; o < 64; ++o) vout[o * 4] = vacc[o];

  float* qo = qT + ((size_t)b * NPIX + n) * 8;
  float* ko = kT + ((size_t)b * NPIX + n) * 8;
  v4f t;
#pragma unroll
  for (int i = 0; i < 4; ++i) t[i] = qacc[i];
  *(v4f*)qo = t;
#pragma unroll
  for (int i = 0; i < 4; ++i) t[i] = qacc[4 + i];
  *(v4f*)(qo + 4) = t;
#pragma unroll
  for (int i = 0; i < 4; ++i) t[i] = kacc[i];
  *(v4f*)ko = t;
#pragma unroll
  for (int i = 0; i < 4; ++i) t[i] = kacc[4 + i];
  *(v4f*)(ko + 4) = t;
}

// ---------------------------------------------------------------------------
// Kernel 2: fused scores -> batch-axis softmax -> out = V @ W.
// One workgroup (8 waves, wave w = batch w) per 16-wide m-tile.
// grid = 256 blocks of 256 threads.
// ---------------------------------------------------------------------------
__global__ __launch_bounds__(256) void attn_kernel(
    const float* __restrict__ vA, const float* __restrict__ qT,
    const float* __restrict__ kT, float* __restrict__ out)
{
  __shared__ float e_lds[8 * 256];   // exp(scores): [batch][n'(16)][m'(16)]
  __shared__ float rcp_lds[256];     // 1 / sum_b exp(scores)

  const int tid  = threadIdx.x;
  const int b    = tid >> 5;         // wave == batch
  const int lane = tid & 31;
  const int lh   = lane & 15;
  const int hi   = lane >> 4;        // 0 or 1
  const int m0   = blockIdx.x << 4;

  // K B-operands (loop-invariant): B(4x16) lane layout — lanes 0-15: K=0,1 ;
  // lanes 16-31: K=2,3.   kb0 -> c=0..3 slice, kb1 -> c=4..7 slice.
  const float* krow = kT + ((size_t)b * NPIX + (m0 + lh)) * 8;
  const v2f kb0 = *(const v2f*)&krow[2 * hi];
  const v2f kb1 = *(const v2f*)&krow[4 + 2 * hi];

  const v8f vzero = {0.f, 0.f, 0.f, 0.f, 0.f, 0.f, 0.f, 0.f};
  v8f acc[4];
#pragma unroll
  for (int cc = 0; cc < 4; ++cc) acc[cc] = vzero;

  const float* qTb = qT + (size_t)b * NPIX * 8;
  const float* vAb = vA + (size_t)b * NBLK * 256;
  float* eb = &e_lds[b * 256];

  for (int step = 0; step < 256; ++step) {
    const int n0 = step << 4;

    // ---- scores S(16n x 16m) = Qt(16x8) @ K(8x16), two K=4 WMMAs ----
    const float* qrow = qTb + (size_t)(n0 + lh) * 8;
    v2f qa0 = *(const v2f*)&qrow[2 * hi];
    v2f qa1 = *(const v2f*)&qrow[4 + 2 * hi];
    v8f s = wmma4(qa0, kb0, vzero);
    s = wmma4(qa1, kb1, s);

    // exp in-register, stash to LDS (C layout: row = r + 8*hi, col = lh)
#pragma unroll
    for (int r = 0; r < 8; ++r) {
      float e = __builtin_amdgcn_exp2f(s[r] * LOG2E);
      eb[(r + 8 * hi) * 16 + lh] = e;
    }
    __syncthreads();

    // ---- softmax over batch axis: each thread owns one (n',m') element ----
    float den = 0.f;
#pragma unroll
    for (int bb = 0; bb < 8; ++bb) den += e_lds[bb * 256 + tid];
    rcp_lds[tid] = 1.0f / den;
    __syncthreads();

    // ---- weight B-operands (4 K-slices of 4 n-rows each) ----
    v2f wb[4];
#pragma unroll
    for (int j = 0; j < 4; ++j) {
      int r0 = 4 * j + 2 * hi;
      wb[j][0] = eb[r0 * 16 + lh]       * rcp_lds[r0 * 16 + lh];
      wb[j][1] = eb[(r0 + 1) * 16 + lh] * rcp_lds[(r0 + 1) * 16 + lh];
    }

    // ---- out-tile accumulate: 4 c-chunks x 4 K-slices = 16 WMMAs ----
#pragma unroll
    for (int j = 0; j < 4; ++j) {
      const float* vblk = vAb + ((size_t)(n0 >> 2) + j) * 256;
#pragma unroll
      for (int cc = 0; cc < 4; ++cc) {
        v2f va = *(const v2f*)&vblk[(cc * 16 + lh) * 4 + 2 * hi];
        acc[cc] = wmma4(va, wb[j], acc[cc]);
      }
    }
    __syncthreads();  // protect e_lds/rcp_lds before next step overwrites
  }

  // ---- epilogue: D layout -> out[b][c][m], c = cc*16 + r + 8*hi, m = m0+lh
  float* ob = out + (size_t)b * 64 * NPIX + m0 + lh;
#pragma unroll
  for (int cc = 0; cc < 4; ++cc)
#pragma unroll
    for (int r = 0; r < 8; ++r)
      ob[(size_t)(cc * 16 + r + 8 * hi) * NPIX] = acc[cc][r];
}

// ---------------------------------------------------------------------------
extern "C" void kernel_launch(void* const* d_in, const int* in_sizes, int n_in,
                              void* d_out, int out_size, void* d_ws, size_t ws_size,
                              hipStream_t stream) {
  const float* x  = (const float*)d_in[0];
  const float* W1 = (const float*)d_in[1];
  const float* b1 = (const float*)d_in[2];
  const float* W2 = (const float*)d_in[3];
  const float* b2 = (const float*)d_in[4];
  const float* W3 = (const float*)d_in[5];
  const float* b3 = (const float*)d_in[6];
  float* out = (float*)d_out;

  // workspace: vA (8 MiB) | qT (1 MiB) | kT (1 MiB)
  float* vA = (float*)d_ws;
  float* qT = vA + (size_t)8 * NBLK * 256;   // 2,097,152 floats
  float* kT = qT + (size_t)8 * NPIX * 8;     //   262,144 floats

  proj_kernel<<<128, 256, 0, stream>>>(x, W1, b1, W2, b2, W3, b3, vA, qT, kT);
  attn_kernel<<<256, 256, 0, stream>>>(vA, qT, kT, out);
}